// RoIHeads_74242804679515
// MI455X (gfx1250) — compile-verified
//
#include <hip/hip_runtime.h>
#include <math.h>

#define NPROP 2048
#define NCLS  21
#define CF    20
#define NDET  100
#define SCORE_TH 0.05f
#define NMS_TH   0.5f

typedef __attribute__((ext_vector_type(2))) float v2f;
typedef __attribute__((ext_vector_type(8))) float v8f;
typedef int v4i_ __attribute__((vector_size(4 * sizeof(int))));

#define AS1 __attribute__((address_space(1)))
#define AS3 __attribute__((address_space(3)))

#if defined(__has_builtin)
# if __has_builtin(__builtin_amdgcn_global_load_async_to_lds_b128)
#  define HAVE_ASYNC_LDS 1
# endif
#endif
#ifndef HAVE_ASYNC_LDS
# define HAVE_ASYNC_LDS 0
#endif

__device__ __forceinline__ void wait_asynccnt0() {
#if defined(__has_builtin) && __has_builtin(__builtin_amdgcn_s_wait_asynccnt)
    __builtin_amdgcn_s_wait_asynccnt(0);
#else
    asm volatile("s_wait_asynccnt 0" ::: "memory");
#endif
}

// ---------------------------------------------------------------------------
// Kernel 1: softmax + box decode + clip + validity, class-major outputs.
// scoresT: [CF][NPROP]  (invalid -> -1),  boxesT: [CF][NPROP][4]
// ---------------------------------------------------------------------------
__global__ __launch_bounds__(256)
void k_decode(const float* __restrict__ logit, const float* __restrict__ reg,
              const float* __restrict__ prop,  const int* __restrict__ imgsz,
              float* __restrict__ scoresT, float* __restrict__ boxesT)
{
    int n = blockIdx.x * blockDim.x + threadIdx.x;
    if (n >= NPROP) return;
    float img = (float)imgsz[0];

    float l[NCLS];
    float m = -3.0e38f;
#pragma unroll
    for (int c = 0; c < NCLS; ++c) { l[c] = logit[n * NCLS + c]; m = fmaxf(m, l[c]); }
    float sum = 0.f;
#pragma unroll
    for (int c = 0; c < NCLS; ++c) { l[c] = __expf(l[c] - m); sum += l[c]; }
    float inv = 1.0f / sum;

    float x1 = prop[n*4+0], y1 = prop[n*4+1], x2 = prop[n*4+2], y2 = prop[n*4+3];
    float w  = x2 - x1, h = y2 - y1;
    float cx = x1 + 0.5f * w, cy = y1 + 0.5f * h;
    const float clipv = 4.135166556742356f;  // log(1000/16)

    for (int c = 1; c < NCLS; ++c) {
        const float* d = &reg[(size_t)n * NCLS * 4 + c * 4];
        float dx = d[0] * 0.1f;                    // / WX
        float dy = d[1] * 0.1f;                    // / WY
        float dw = fminf(d[2] * 0.2f, clipv);      // / WW, clipped
        float dh = fminf(d[3] * 0.2f, clipv);      // / WH, clipped
        float pcx = dx * w + cx, pcy = dy * h + cy;
        float pw  = __expf(dw) * w, ph = __expf(dh) * h;
        float bx1 = fminf(fmaxf(pcx - 0.5f * pw, 0.f), img);
        float by1 = fminf(fmaxf(pcy - 0.5f * ph, 0.f), img);
        float bx2 = fminf(fmaxf(pcx + 0.5f * pw, 0.f), img);
        float by2 = fminf(fmaxf(pcy + 0.5f * ph, 0.f), img);
        float score = l[c] * inv;
        bool valid = (score >= SCORE_TH) && ((bx2 - bx1) >= 1.0f) && ((by2 - by1) >= 1.0f);
        int ci = c - 1;
        scoresT[ci * NPROP + n] = valid ? score : -1.0f;
        float* bo = &boxesT[((size_t)ci * NPROP + n) * 4];
        bo[0] = bx1; bo[1] = by1; bo[2] = bx2; bo[3] = by2;
    }
}

// ---------------------------------------------------------------------------
// Kernel 2: per-class bitonic sort (desc by score, ties by index asc) in LDS.
// Scores staged global->LDS with async-to-LDS (ASYNCcnt) when available.
// ---------------------------------------------------------------------------
__device__ __forceinline__ bool before_(float sa, int ia, float sb, int ib) {
    return (sa > sb) || (sa == sb && ia < ib);
}

__global__ __launch_bounds__(512)
void k_sort(const float* __restrict__ scoresT, const float* __restrict__ boxesT,
            float* __restrict__ sortScore, float* __restrict__ sortBox)
{
    __shared__ float ss[NPROP];
    __shared__ int   si[NPROP];
    int c = blockIdx.x;
    const float* s = &scoresT[(size_t)c * NPROP];

#if HAVE_ASYNC_LDS
    // 4 floats (16B) per thread: 512 threads cover 2048 scores.
    for (int t = threadIdx.x * 4; t < NPROP; t += blockDim.x * 4) {
        __builtin_amdgcn_global_load_async_to_lds_b128(
            (AS1 v4i_*)(s + t), (AS3 v4i_*)(&ss[t]), 0, 0);
    }
    for (int t = threadIdx.x; t < NPROP; t += blockDim.x) si[t] = t;
    wait_asynccnt0();          // per-wave: own async writes landed in LDS
    __syncthreads();           // cross-wave ordering
#else
    for (int t = threadIdx.x; t < NPROP; t += blockDim.x) { ss[t] = s[t]; si[t] = t; }
    __syncthreads();
#endif

    for (int k = 2; k <= NPROP; k <<= 1) {
        for (int j = k >> 1; j > 0; j >>= 1) {
            for (int t = threadIdx.x; t < NPROP; t += blockDim.x) {
                int ixj = t ^ j;
                if (ixj > t) {
                    bool dir = ((t & k) == 0);  // ascending in "before" order
                    float s0 = ss[t], s1 = ss[ixj];
                    int   i0 = si[t], i1 = si[ixj];
                    if (before_(s1, i1, s0, i0) == dir) {
                        ss[t] = s1; ss[ixj] = s0;
                        si[t] = i1; si[ixj] = i0;
                    }
                }
            }
            __syncthreads();
        }
    }

    for (int t = threadIdx.x; t < NPROP; t += blockDim.x) {
        sortScore[(size_t)c * NPROP + t] = ss[t];
        int idx = si[t];
        const float* b = &boxesT[((size_t)c * NPROP + idx) * 4];
        float* o = &sortBox[((size_t)c * NPROP + t) * 4];
        o[0] = b[0]; o[1] = b[1]; o[2] = b[2]; o[3] = b[3];
    }
}

// ---------------------------------------------------------------------------
// Kernel 3: suppression bitmask. One wave32 per (class, 16-row tile, 64-col
// word). Area-sum term of the IoU union via v_wmma_f32_16x16x4_f32:
//   A[i][:] = [area_row_i, 1, 0, 0],  B[:][j] = [1, area_col_j, 0, 0]^T
//   => D[i][j] = area_i + area_j.  Bits assembled with wave32 ballots.
// Row boxes staged global->LDS with async-to-LDS when available.
// masks: [CF][NPROP][32] u64
// ---------------------------------------------------------------------------
__global__ __launch_bounds__(32)
void k_mask(const float* __restrict__ sortBox, unsigned long long* __restrict__ masks)
{
    int w    = blockIdx.x;        // 64-col word index, 0..31
    int ti   = blockIdx.y;        // 16-row tile, 0..127
    int c    = blockIdx.z;        // class
    int lane = threadIdx.x;
    int i0 = ti * 16, j0 = w * 64;

    __shared__ float rbox[16][4];
    __shared__ float rarea[16];
    const float* B = &sortBox[(size_t)c * NPROP * 4];

#if HAVE_ASYNC_LDS
    if (lane < 16) {
        __builtin_amdgcn_global_load_async_to_lds_b128(
            (AS1 v4i_*)(B + (size_t)(i0 + lane) * 4),
            (AS3 v4i_*)(&rbox[lane][0]), 0, 0);
    }
    wait_asynccnt0();
    __syncthreads();
    if (lane < 16) {
        rarea[lane] = (rbox[lane][2] - rbox[lane][0]) * (rbox[lane][3] - rbox[lane][1]);
    }
    __syncthreads();
#else
    if (lane < 16) {
        const float* rb = &B[(size_t)(i0 + lane) * 4];
        float x1 = rb[0], y1 = rb[1], x2 = rb[2], y2 = rb[3];
        rbox[lane][0] = x1; rbox[lane][1] = y1; rbox[lane][2] = x2; rbox[lane][3] = y2;
        rarea[lane] = (x2 - x1) * (y2 - y1);
    }
    __syncthreads();
#endif

    // A fragment: 16x4 f32. VGPR0 holds K=0 (lanes 0-15) / K=2 (lanes 16-31),
    // VGPR1 holds K=1 / K=3. Only K=0,1 are non-zero -> lanes >=16 carry zeros.
    v2f a;
    a.x = (lane < 16) ? rarea[lane] : 0.0f;
    a.y = (lane < 16) ? 1.0f        : 0.0f;

    unsigned long long word = 0ull;
#pragma unroll
    for (int s = 0; s < 4; ++s) {
        int col = j0 + s * 16 + (lane & 15);
        const float* cb = &B[(size_t)col * 4];
        float cx1 = cb[0], cy1 = cb[1], cx2 = cb[2], cy2 = cb[3];
        float carea = (cx2 - cx1) * (cy2 - cy1);

        // B fragment: 4x16 f32, mirrored layout (K=0/2 in VGPR0, K=1/3 in VGPR1).
        v2f b;
        b.x = (lane < 16) ? 1.0f  : 0.0f;   // row K=0 = ones
        b.y = (lane < 16) ? carea : 0.0f;   // row K=1 = col areas

        v8f acc = {};
        acc = __builtin_amdgcn_wmma_f32_16x16x4_f32(
                  false, a, false, b, (short)0, acc, false, false);

        unsigned long long part = 0ull;
#pragma unroll
        for (int r = 0; r < 8; ++r) {
            int M = r + ((lane >= 16) ? 8 : 0);         // row within tile
            float rx1 = rbox[M][0], ry1 = rbox[M][1];
            float rx2 = rbox[M][2], ry2 = rbox[M][3];
            float ix1 = fmaxf(rx1, cx1), iy1 = fmaxf(ry1, cy1);
            float ix2 = fminf(rx2, cx2), iy2 = fminf(ry2, cy2);
            float iw = fmaxf(ix2 - ix1, 0.f), ih = fmaxf(iy2 - iy1, 0.f);
            float inter = iw * ih;
            float uni = fmaxf(acc[r] - inter, 1e-6f);   // area_i + area_j - inter
            bool pred = (inter > NMS_TH * uni);         // iou > TH
            unsigned int bv = __builtin_amdgcn_ballot_w32(pred);
            // ballot: bits[15:0] = row r cols 0-15, bits[31:16] = row r+8 cols 0-15
            if ((lane & 7) == r)
                part = (lane < 8) ? (unsigned long long)(bv & 0xFFFFu)
                                  : (unsigned long long)(bv >> 16);
        }
        word |= part << (16 * s);
    }

    if (lane < 16)
        masks[((size_t)c * NPROP + i0 + lane) * 32 + w] = word;
}

// ---------------------------------------------------------------------------
// Kernel 4: serial greedy scan + compaction. One wave32 per class; lane l
// owns removal word l (32 x u64 = 2048 bits). Writes all outputs.
// out layout: boxes[2000*4] | scores[2000] | labels[2000] | valid[2000]
// ---------------------------------------------------------------------------
__global__ __launch_bounds__(32)
void k_scan(const float* __restrict__ sortScore, const float* __restrict__ sortBox,
            const unsigned long long* __restrict__ masks, float* __restrict__ out)
{
    int c = blockIdx.x;
    int lane = threadIdx.x;

    float* ob = out;               // [CF*NDET][4]
    float* os = out + 8000;        // [CF*NDET]
    float* ol = out + 10000;       // labels
    float* ov = out + 12000;       // valid

    const float* S  = &sortScore[(size_t)c * NPROP];
    const float* Bx = &sortBox[(size_t)c * NPROP * 4];
    const unsigned long long* M = &masks[(size_t)c * NPROP * 32];

    unsigned long long remv = 0ull;
    int cnt = 0;

    for (int i = 0; i < NPROP; ++i) {
        if (i + 1 < NPROP)
            __builtin_prefetch(&M[(size_t)(i + 1) * 32 + lane], 0, 3);  // global_prefetch_b8
        float sc = S[i];                       // uniform across the wave
        if (sc < 0.0f) continue;               // invalid: never kept, never suppresses
        int wrd = i >> 6, bit = i & 63;
        unsigned int rlo = (unsigned int)remv;
        unsigned int rhi = (unsigned int)(remv >> 32);
        unsigned int slo = (unsigned int)__shfl((int)rlo, wrd, 32);
        unsigned int shi = (unsigned int)__shfl((int)rhi, wrd, 32);
        unsigned long long rw = ((unsigned long long)shi << 32) | slo;
        if (!((rw >> bit) & 1ull)) {
            remv |= M[(size_t)i * 32 + lane];  // mark everything this box overlaps
            if (cnt < NDET) {
                int slot = c * NDET + cnt;
                if (lane < 4)  ob[(size_t)slot * 4 + lane] = Bx[(size_t)i * 4 + lane];
                if (lane == 4) os[slot] = sc;
            }
            ++cnt;
        }
    }

    // labels, valid flags, zero-fill for empty slots
    for (int t = lane; t < NDET; t += 32) {
        int slot = c * NDET + t;
        ol[slot] = (float)(c + 1);
        ov[slot] = (t < cnt) ? 1.0f : 0.0f;
        if (t >= cnt) {
            os[slot] = 0.0f;
            ob[(size_t)slot * 4 + 0] = 0.0f;
            ob[(size_t)slot * 4 + 1] = 0.0f;
            ob[(size_t)slot * 4 + 2] = 0.0f;
            ob[(size_t)slot * 4 + 3] = 0.0f;
        }
    }
}

// ---------------------------------------------------------------------------
extern "C" void kernel_launch(void* const* d_in, const int* in_sizes, int n_in,
                              void* d_out, int out_size, void* d_ws, size_t ws_size,
                              hipStream_t stream)
{
    (void)in_sizes; (void)n_in; (void)out_size; (void)ws_size;
    const float* logit = (const float*)d_in[0];
    const float* reg   = (const float*)d_in[1];
    const float* prop  = (const float*)d_in[2];
    const int*   img   = (const int*)d_in[3];
    float* out = (float*)d_out;

    char* ws = (char*)d_ws;
    size_t off = 0;
    auto carve = [&](size_t bytes) -> void* {
        void* p = (void*)(ws + off);
        off += (bytes + 255) & ~(size_t)255;
        return p;
    };
    float* scoresT   = (float*)carve((size_t)CF * NPROP * 4);            // 160 KB
    float* boxesT    = (float*)carve((size_t)CF * NPROP * 4 * 4);        // 640 KB
    float* sortScore = (float*)carve((size_t)CF * NPROP * 4);            // 160 KB
    float* sortBox   = (float*)carve((size_t)CF * NPROP * 4 * 4);        // 640 KB
    unsigned long long* masks =
        (unsigned long long*)carve((size_t)CF * NPROP * 32 * 8);         // 10 MB

    k_decode<<<(NPROP + 255) / 256, 256, 0, stream>>>(logit, reg, prop, img,
                                                      scoresT, boxesT);
    k_sort<<<CF, 512, 0, stream>>>(scoresT, boxesT, sortScore, sortBox);
    dim3 mg(32, NPROP / 16, CF);
    k_mask<<<mg, 32, 0, stream>>>(sortBox, masks);
    k_scan<<<CF, 32, 0, stream>>>(sortScore, sortBox, masks, out);
}